// NoisyTopKRouter_70720931496135
// MI455X (gfx1250) — compile-verified
//
#include <hip/hip_runtime.h>
#include <hip/hip_bf16.h>
#include <math.h>

#define NUM_E 64
#define TOPK  2
#define MINP  0.001f
#define REPS  1e-10f

typedef __attribute__((ext_vector_type(2))) float v2f;
typedef __attribute__((ext_vector_type(4))) float v4f;
typedef __attribute__((ext_vector_type(8))) float v8f;

// -------------------------------------------------------------------------
// Kernel 1: logits[T,64] = hs[T,H] @ W_gate[64,H]^T  via V_WMMA_F32_16X16X4_F32
// Block = 128 threads = 4 waves. Block b owns tokens [16b,16b+16); wave j owns
// expert columns [16j,16j+16). A-tile per ISA layout: lane<16 holds row M=lane
// (K=k0,k0+1), lane>=16 holds row M=lane-16 (K=k0+2,k0+3) -> float2 loads.
// B-tile (K x N on W_gate rows) mirrors the same +2*(lane>=16) pattern.
// -------------------------------------------------------------------------
__global__ __launch_bounds__(128) void moe_logits_wmma(
    const float* __restrict__ hs, const float* __restrict__ wg,
    float* __restrict__ logits, int T, int H)
{
  const int lane  = threadIdx.x & 31;
  const int etile = threadIdx.x >> 5;   // 0..3
  const int half  = lane >> 4;          // 0 or 1
  const int l15   = lane & 15;
  const size_t tb = (size_t)blockIdx.x * 16;

  const float* ap = hs + (tb + (size_t)l15) * H + 2 * half;
  const float* bp = wg + ((size_t)etile * 16 + l15) * H + 2 * half;

  v8f c0 = {};
  v8f c1 = {};
  int k0 = 0;
  for (; k0 + 8 <= H; k0 += 8) {
    v2f a0 = *(const v2f*)(ap + k0);
    v2f b0 = *(const v2f*)(bp + k0);
    v2f a1 = *(const v2f*)(ap + k0 + 4);
    v2f b1 = *(const v2f*)(bp + k0 + 4);
    c0 = __builtin_amdgcn_wmma_f32_16x16x4_f32(false, a0, false, b0, (short)0, c0, false, false);
    c1 = __builtin_amdgcn_wmma_f32_16x16x4_f32(false, a1, false, b1, (short)0, c1, false, false);
  }
  for (; k0 < H; k0 += 4) {
    v2f a0 = *(const v2f*)(ap + k0);
    v2f b0 = *(const v2f*)(bp + k0);
    c0 = __builtin_amdgcn_wmma_f32_16x16x4_f32(false, a0, false, b0, (short)0, c0, false, false);
  }
  v8f c = c0 + c1;

  const int col = etile * 16 + l15;     // expert index
#pragma unroll
  for (int v = 0; v < 8; ++v) {
    // D layout: VGPR v holds M = v + 8*half, N = lane%16
    logits[(tb + (size_t)v + 8 * half) * NUM_E + col] = c[v];
  }
}

// -------------------------------------------------------------------------
// Kernel 2: per-token softmax + min-prob mix + renorm + top-2 + dense outputs
// One wave32 per token, 8 tokens per 256-thread block. Deterministic per-block
// partial sums for the aux-loss reductions.
// -------------------------------------------------------------------------
__device__ __forceinline__ bool viGT(float v0, int i0, float v1, int i1) {
  return (v0 > v1) || (v0 == v1 && i0 < i1);
}

__global__ __launch_bounds__(256) void moe_router(
    const float* __restrict__ logits,
    float* __restrict__ mask_out, float* __restrict__ wt_out,
    float* __restrict__ gate_part, float* __restrict__ usage_part,
    float* __restrict__ ent_part, float* __restrict__ z_part)
{
  __shared__ float sc_lds[8][NUM_E];
  __shared__ int   ia_lds[8], ib_lds[8];
  __shared__ float ez_lds[8][2];

  const int lane = threadIdx.x & 31;
  const int wv   = threadIdx.x >> 5;
  const size_t t = (size_t)blockIdx.x * 8 + wv;

  float x0 = logits[t * NUM_E + lane];
  float x1 = logits[t * NUM_E + 32 + lane];

  // softmax over 64 entries (2 per lane)
  float mx = fmaxf(x0, x1);
  for (int m = 16; m >= 1; m >>= 1) mx = fmaxf(mx, __shfl_xor(mx, m, 32));
  float e0 = expf(x0 - mx), e1 = expf(x1 - mx);
  float s = e0 + e1;
  for (int m = 16; m >= 1; m >>= 1) s += __shfl_xor(s, m, 32);
  float inv = 1.0f / s;

  const float mixw = 1.0f - MINP * (float)NUM_E;
  float sc0 = (e0 * inv) * mixw + MINP;
  float sc1 = (e1 * inv) * mixw + MINP;
  float s2 = sc0 + sc1;
  for (int m = 16; m >= 1; m >>= 1) s2 += __shfl_xor(s2, m, 32);
  float inv2 = 1.0f / s2;
  sc0 *= inv2; sc1 *= inv2;

  // top-2 butterfly (keeps a sorted pair per lane; tie -> lower index)
  float av, bv; int ai, bi;
  if (sc1 > sc0) { av = sc1; ai = lane + 32; bv = sc0; bi = lane; }
  else           { av = sc0; ai = lane;      bv = sc1; bi = lane + 32; }
  for (int m = 16; m >= 1; m >>= 1) {
    float av2 = __shfl_xor(av, m, 32);
    int   ai2 = __shfl_xor(ai, m, 32);
    float bv2 = __shfl_xor(bv, m, 32);
    int   bi2 = __shfl_xor(bi, m, 32);
    if (viGT(av, ai, av2, ai2)) {
      if (!viGT(bv, bi, av2, ai2)) { bv = av2; bi = ai2; }
    } else {
      if (viGT(bv2, bi2, av, ai)) { bv = bv2; bi = bi2; }
      else                        { bv = av;  bi = ai; }
      av = av2; ai = ai2;
    }
  }
  float wsum = av + bv;
  float wa = av / wsum, wb = bv / wsum;

  // dense [E,K] outputs: each lane owns 4 consecutive slots (no races)
  const size_t base = t * (NUM_E * TOPK);
  v4f mvec, wvec;
#pragma unroll
  for (int i = 0; i < 4; ++i) {
    int pos = lane * 4 + i;
    int e = pos >> 1, k = pos & 1;
    bool hit = (k == 0) ? (e == ai) : (e == bi);
    mvec[i] = hit ? 1.0f : 0.0f;
    wvec[i] = hit ? (k == 0 ? wa : wb) : 0.0f;
  }
  *(v4f*)(mask_out + base + (size_t)lane * 4) = mvec;
  *(v4f*)(wt_out   + base + (size_t)lane * 4) = wvec;

  // per-token entropy and z terms
  float entl = sc0 * logf(sc0 + REPS) + sc1 * logf(sc1 + REPS);
  for (int m = 16; m >= 1; m >>= 1) entl += __shfl_xor(entl, m, 32);
  float zl = expf(sc0) + expf(sc1);
  for (int m = 16; m >= 1; m >>= 1) zl += __shfl_xor(zl, m, 32);

  sc_lds[wv][lane]      = sc0;
  sc_lds[wv][lane + 32] = sc1;
  if (lane == 0) {
    ia_lds[wv] = ai; ib_lds[wv] = bi;
    ez_lds[wv][0] = -entl;
    float lse = logf(zl);
    ez_lds[wv][1] = lse * lse;
  }
  __syncthreads();

  const int tid = threadIdx.x;
  if (tid < NUM_E) {
    float g = 0.0f; int u = 0;
#pragma unroll
    for (int w = 0; w < 8; ++w) {
      g += sc_lds[w][tid];
      u += (ia_lds[w] == tid) + (ib_lds[w] == tid);
    }
    gate_part [(size_t)blockIdx.x * NUM_E + tid] = g;
    usage_part[(size_t)blockIdx.x * NUM_E + tid] = (float)u;
  } else if (tid == NUM_E) {
    float es = 0.0f, zs = 0.0f;
#pragma unroll
    for (int w = 0; w < 8; ++w) { es += ez_lds[w][0]; zs += ez_lds[w][1]; }
    ent_part[blockIdx.x] = es;
    z_part  [blockIdx.x] = zs;
  }
}

// -------------------------------------------------------------------------
// Kernel 3: single block reduces the per-block partials and emits scalars.
// -------------------------------------------------------------------------
__global__ __launch_bounds__(256) void moe_finalize(
    const float* __restrict__ gate_part, const float* __restrict__ usage_part,
    const float* __restrict__ ent_part, const float* __restrict__ z_part,
    float* __restrict__ out_tail, int T, int NB)
{
  __shared__ float gp[NUM_E], us[NUM_E];
  __shared__ float red[256];
  const int tid = threadIdx.x;

  if (tid < NUM_E) {
    float g = 0.0f, u = 0.0f;
    for (int b = 0; b < NB; ++b) {
      g += gate_part [(size_t)b * NUM_E + tid];
      u += usage_part[(size_t)b * NUM_E + tid];
    }
    gp[tid] = g / (float)T;   // mean over tokens
    us[tid] = u;
  }

  float es = 0.0f, zs = 0.0f;
  for (int b = tid; b < NB; b += 256) { es += ent_part[b]; zs += z_part[b]; }
  red[tid] = es;
  for (int o = 128; o >= 1; o >>= 1) { __syncthreads(); if (tid < o) red[tid] += red[tid + o]; }
  __syncthreads();
  float ent_total = red[0];
  __syncthreads();
  red[tid] = zs;
  for (int o = 128; o >= 1; o >>= 1) { __syncthreads(); if (tid < o) red[tid] += red[tid + o]; }
  __syncthreads();
  float z_total = red[0];

  if (tid == 0) {
    float total = REPS;
    for (int e = 0; e < NUM_E; ++e) total += us[e];
    float load = 0.0f, gsum = 0.0f, eent = 0.0f;
    for (int e = 0; e < NUM_E; ++e) {
      float un = us[e] / total;
      out_tail[4 + e] = un;                       // expert_usage_normalized
      load += gp[e] * un;
      gsum += gp[e];
      eent -= un * logf(un + REPS);
    }
    float mean = gsum / (float)NUM_E;
    float var = 0.0f;
    for (int e = 0; e < NUM_E; ++e) { float d = gp[e] - mean; var += d * d; }
    var /= (float)NUM_E;

    out_tail[0] = fminf(fmaxf(load * (float)NUM_E, 0.0f), 10.0f);        // load_loss
    out_tail[1] = fminf(fmaxf(var / (mean * mean + REPS), 0.0f), 1.0f);  // importance_loss
    out_tail[2] = fminf(fmaxf(z_total / (float)T, 0.0f), 100.0f);        // z_loss
    float rent = ent_total / (float)T;
    out_tail[3] = fminf(fmaxf((1.0f - rent / logf((float)NUM_E)) * 5.0f, 0.0f), 5.0f); // entropy_reg
    out_tail[4 + NUM_E] = eent;                                          // expert_entropy
  }
}

// -------------------------------------------------------------------------
extern "C" void kernel_launch(void* const* d_in, const int* in_sizes, int n_in,
                              void* d_out, int out_size, void* d_ws, size_t ws_size,
                              hipStream_t stream)
{
  (void)n_in; (void)out_size; (void)ws_size;
  const float* hs = (const float*)d_in[0];
  const float* wg = (const float*)d_in[1];
  float* out = (float*)d_out;

  const int H = in_sizes[1] / NUM_E;                 // 2048
  const long long T = (long long)in_sizes[0] / H;    // 16384
  const int NB = (int)(T / 8);

  float* logits     = (float*)d_ws;                  // T*64
  float* gate_part  = logits + (size_t)T * NUM_E;    // NB*64
  float* usage_part = gate_part + (size_t)NB * NUM_E;
  float* ent_part   = usage_part + (size_t)NB * NUM_E;
  float* z_part     = ent_part + NB;

  float* mask_out = out;
  float* wt_out   = out + (size_t)T * NUM_E * TOPK;
  float* out_tail = out + 2 * (size_t)T * NUM_E * TOPK;

  moe_logits_wmma<<<dim3((unsigned)(T / 16)), dim3(128), 0, stream>>>(hs, wg, logits, (int)T, H);
  moe_router<<<dim3((unsigned)NB), dim3(256), 0, stream>>>(logits, mask_out, wt_out,
                                                           gate_part, usage_part, ent_part, z_part);
  moe_finalize<<<dim3(1), dim3(256), 0, stream>>>(gate_part, usage_part, ent_part, z_part,
                                                  out_tail, (int)T, NB);
}